// ScanCompensation_1297080123622
// MI455X (gfx1250) — compile-verified
//
#include <hip/hip_runtime.h>
#include <hip/hip_bf16.h>

#define HDIM 480
#define WDIM 640
#define PIX (HDIM * WDIM)   // 307200 pixels per bin
#define SPLIT 8             // blocks per bin in the variance pass

typedef __attribute__((ext_vector_type(2))) float v2f;
typedef __attribute__((ext_vector_type(8))) float v8f;

// ---- total-order float <-> uint encoding (for atomic min on floats) ----
__device__ __forceinline__ unsigned floatToOrderedU(float f) {
    unsigned u = __float_as_uint(f);
    return (u & 0x80000000u) ? ~u : (u | 0x80000000u);
}
__device__ __forceinline__ float orderedUToFloat(unsigned e) {
    unsigned bits = (e & 0x80000000u) ? (e & 0x7FFFFFFFu) : ~e;
    return __uint_as_float(bits);
}

// ---------------- phase 0: clear output (tensor + loss slot) ----------------
__global__ __launch_bounds__(256) void zero_out(float* __restrict__ p, int n) {
    int i = blockIdx.x * blockDim.x + threadIdx.x;
    int stride = gridDim.x * blockDim.x;
    int n4 = n >> 2;
    float4* p4 = reinterpret_cast<float4*>(p);
    float4 z = make_float4(0.f, 0.f, 0.f, 0.f);
    for (int k = i; k < n4; k += stride) p4[k] = z;
    for (int k = (n4 << 2) + i; k < n; k += stride) p[k] = 0.f;
}

// ---------------- phase 0b: reset workspace accumulators ----------------
__global__ void init_ws(unsigned* __restrict__ wsu, float* __restrict__ wsf, int nb) {
    if (threadIdx.x == 0) wsu[0] = 0xFFFFFFFFu;           // encoded +inf (min identity)
    for (int k = threadIdx.x; k < 2 * nb; k += blockDim.x) wsf[k] = 0.f;
}

// ---------------- phase 1: min over warped timestamps ----------------
__global__ __launch_bounds__(256) void min_kernel(const float* __restrict__ params,
                                                  const float* __restrict__ x,
                                                  const float* __restrict__ y,
                                                  const float* __restrict__ ts,
                                                  int n, unsigned* __restrict__ wsu) {
    float a0 = params[0], a1 = params[1];
    int i = blockIdx.x * blockDim.x + threadIdx.x;
    int stride = gridDim.x * blockDim.x;
    float m = 3.4e38f;
    for (int k = i; k < n; k += stride) {
        float t = ts[k] - a0 * x[k] - a1 * y[k];
        m = fminf(m, t);
    }
    for (int off = 16; off; off >>= 1) m = fminf(m, __shfl_xor(m, off, 32));
    __shared__ float sm[8];
    int lane = threadIdx.x & 31, wave = threadIdx.x >> 5;
    if (lane == 0) sm[wave] = m;
    __syncthreads();
    if (threadIdx.x == 0) {
        float bm = sm[0];
        int nw = blockDim.x >> 5;
        for (int w = 1; w < nw; ++w) bm = fminf(bm, sm[w]);
        atomicMin(wsu, floatToOrderedU(bm));   // global_atomic_min_u32
    }
}

// ---------------- phase 2: bilinear-in-time scatter-add ----------------
__global__ __launch_bounds__(256) void scatter_kernel(const float* __restrict__ params,
                                                      const float* __restrict__ x,
                                                      const float* __restrict__ y,
                                                      const float* __restrict__ ts,
                                                      const float* __restrict__ pol,
                                                      const float* __restrict__ bwp,
                                                      int n, int nb,
                                                      const unsigned* __restrict__ wsu,
                                                      float* __restrict__ out) {
    float a0 = params[0], a1 = params[1];
    float bw = bwp[0];
    float t_start = orderedUToFloat(wsu[0]);
    int i = blockIdx.x * blockDim.x + threadIdx.x;
    int stride = gridDim.x * blockDim.x;
    for (int k = i; k < n; k += stride) {
        float xf = x[k], yf = y[k];
        float t  = ts[k] - a0 * xf - a1 * yf;
        float tn = (t - t_start) / bw;
        float t0 = floorf(tn);
        float wt = tn - t0;
        int t0c = min(max((int)t0, 0), nb - 1);
        int t1c = min(max((int)(t0 + 1.0f), 0), nb - 1);
        int xi = (int)xf, yi = (int)yf;
        if (xi >= 0 && xi < WDIM && yi >= 0 && yi < HDIM) {
            int spatial = yi * WDIM + xi;
            float p = pol[k];
            atomicAdd(&out[t0c * PIX + spatial], (1.0f - wt) * p);  // global_atomic_add_f32 (L2-resident)
            atomicAdd(&out[t1c * PIX + spatial], wt * p);
        }
    }
}

// ------- phase 3: per-bin sum / sum-of-squares via f32 WMMA reduction -------
// V_WMMA_F32_16X16X4_F32 with B = ones: D[m][n] += sum_k A[m][k].
// 64 f32 values folded into the accumulator tile per instruction, full f32
// precision. Every tile entry in column n holds identical row-sums, so the
// full-tile sum equals 16x the true sum -> exact scale-back by 1/16.
__global__ __launch_bounds__(256) void binvar_kernel(const float* __restrict__ out,
                                                     float* __restrict__ wsf) {
    int bin  = blockIdx.x / SPLIT;
    int part = blockIdx.x % SPLIT;
    const int CHUNK = PIX / SPLIT;          // 38400, divisible by 512
    const float* base = out + (size_t)bin * PIX + (size_t)part * CHUNK;
    int lane = threadIdx.x & 31, wave = threadIdx.x >> 5;  // 8 waves of 32

    v8f cs = {};  // accumulates sum
    v8f cq = {};  // accumulates sum of squares
    v2f ones = {1.0f, 1.0f};

    for (int it = 0; it < CHUNK; it += 256 * 2) {          // 512 elems/block/iter
        int idx = it + wave * 64 + lane * 2;
        v2f a = *(const v2f*)(base + idx);
        v2f a2; a2[0] = a[0] * a[0]; a2[1] = a[1] * a[1];
        cs = __builtin_amdgcn_wmma_f32_16x16x4_f32(false, a,  false, ones, (short)0, cs, false, false);
        cq = __builtin_amdgcn_wmma_f32_16x16x4_f32(false, a2, false, ones, (short)0, cq, false, false);
    }

    float s = 0.f, q = 0.f;
#pragma unroll
    for (int r = 0; r < 8; ++r) { s += cs[r]; q += cq[r]; }
    for (int off = 16; off; off >>= 1) {
        s += __shfl_xor(s, off, 32);
        q += __shfl_xor(q, off, 32);
    }
    s *= (1.0f / 16.0f);
    q *= (1.0f / 16.0f);

    __shared__ float ss[8], sq[8];
    if (lane == 0) { ss[wave] = s; sq[wave] = q; }
    __syncthreads();
    if (threadIdx.x == 0) {
        float S = 0.f, Q = 0.f;
        for (int w = 0; w < 8; ++w) { S += ss[w]; Q += sq[w]; }
        atomicAdd(&wsf[2 * bin],     S);
        atomicAdd(&wsf[2 * bin + 1], Q);
    }
}

// ---------------- phase 4: unbiased variance per bin, mean -> loss ----------------
__global__ __launch_bounds__(256) void finalize_kernel(const float* __restrict__ wsf,
                                                       float* __restrict__ out, int nb) {
    __shared__ float acc[256];
    float v = 0.f;
    const float n = (float)PIX;
    for (int b = threadIdx.x; b < nb; b += blockDim.x) {
        float S = wsf[2 * b], Q = wsf[2 * b + 1];
        v += (Q - S * S / n) / (n - 1.0f);   // torch.var, ddof=1
    }
    acc[threadIdx.x] = v;
    __syncthreads();
    for (int off = 128; off; off >>= 1) {
        if (threadIdx.x < off) acc[threadIdx.x] += acc[threadIdx.x + off];
        __syncthreads();
    }
    if (threadIdx.x == 0) out[(size_t)nb * PIX] = acc[0] / (float)nb;
}

extern "C" void kernel_launch(void* const* d_in, const int* in_sizes, int n_in,
                              void* d_out, int out_size, void* d_ws, size_t ws_size,
                              hipStream_t stream) {
    const float* params = (const float*)d_in[0];
    const float* x      = (const float*)d_in[1];
    const float* y      = (const float*)d_in[2];
    const float* ts     = (const float*)d_in[3];
    const float* pol    = (const float*)d_in[4];
    const float* bw     = (const float*)d_in[7];   // bin_width scalar

    int n  = in_sizes[1];                    // N_EVENTS
    int nb = (out_size - 1) / PIX;           // num_bins recovered from output size
    if (nb < 1) nb = 1;

    float*    out = (float*)d_out;
    unsigned* wsu = (unsigned*)d_ws;         // [0]: encoded running min
    float*    wsf = (float*)d_ws + 8;        // 2*nb floats: (sum, sumsq) per bin

    zero_out       <<<2048, 256, 0, stream>>>(out, out_size);
    init_ws        <<<1,    256, 0, stream>>>(wsu, wsf, nb);
    min_kernel     <<<2048, 256, 0, stream>>>(params, x, y, ts, n, wsu);
    scatter_kernel <<<4096, 256, 0, stream>>>(params, x, y, ts, pol, bw, n, nb, wsu, out);
    binvar_kernel  <<<nb * SPLIT, 256, 0, stream>>>(out, wsf);
    finalize_kernel<<<1,    256, 0, stream>>>(wsf, out, nb);
}